// GCN_72971494359045
// MI455X (gfx1250) — compile-verified
//
#include <hip/hip_runtime.h>

typedef __attribute__((ext_vector_type(2))) float v2f;
typedef __attribute__((ext_vector_type(8))) float v8f;

// ---------------------------------------------------------------------------
// Degree / normalization
// ---------------------------------------------------------------------------
__global__ void deg_init(float* __restrict__ deg, int n) {
  int i = blockIdx.x * blockDim.x + threadIdx.x;
  if (i < n) deg[i] = 1.0f;              // self loop
}

__global__ void deg_edges(const long long* __restrict__ dst,
                          float* __restrict__ deg, int E) {
  int e = blockIdx.x * blockDim.x + threadIdx.x;
  if (e < E) atomicAdd(&deg[(int)dst[e]], 1.0f);
}

__global__ void dinv_k(const float* __restrict__ deg,
                       float* __restrict__ dinv, int n) {
  int i = blockIdx.x * blockDim.x + threadIdx.x;
  if (i < n) dinv[i] = rsqrtf(deg[i]);   // deg >= 1 always
}

// c1[j] = sum_t id[t] * W1[(128+t)*256 + j]  (constant contribution of the
// 8 broadcast "id" feature columns)
__global__ void c1_kernel(const float* __restrict__ idv,
                          const float* __restrict__ W1,
                          float* __restrict__ c1) {
  int j = threadIdx.x;                   // 256 threads
  float s = 0.0f;
#pragma unroll
  for (int t = 0; t < 8; ++t) s += idv[t] * W1[(128 + t) * 256 + j];
  c1[j] = s;
}

// ---------------------------------------------------------------------------
// fp32 WMMA GEMM:  C[M x N] = A[M x K] @ B[K x N] (+ rowvec[N] if non-null)
// lda = K, ldb = ldc = N.  Requires: M % 16 == 0, K % 4 == 0, N % 64 == 0.
// Block = 256 threads (8 waves); each wave owns a 16(M) x 64(N) tile.
// ---------------------------------------------------------------------------
__global__ __launch_bounds__(256) void gemm_f32_wmma(
    const float* __restrict__ A, const float* __restrict__ B,
    const float* __restrict__ rowvec, float* __restrict__ C,
    int M, int N, int K) {
  const int lane  = threadIdx.x & 31;
  const int wave  = threadIdx.x >> 5;
  const int tileM = blockIdx.x * 128 + wave * 16;
  const int tileN = blockIdx.y * 64;
  if (tileM >= M) return;                // wave-uniform: EXEC stays all-1s

  const int mrow  = tileM + (lane & 15); // A row this lane supplies
  const int khalf = (lane >> 4) * 2;     // K pair selector (lanes 16-31 -> +2)
  const int ncol0 = tileN + (lane & 15); // B/C column this lane supplies

  v8f acc[4] = {};

  for (int k = 0; k < K; k += 4) {
    const int kk = k + khalf;
    v2f a;
    a.x = A[(size_t)mrow * K + kk];
    a.y = A[(size_t)mrow * K + kk + 1];
#pragma unroll
    for (int s = 0; s < 4; ++s) {
      const int n = ncol0 + s * 16;
      v2f b;
      b.x = B[(size_t)kk * N + n];
      b.y = B[(size_t)(kk + 1) * N + n];
      acc[s] = __builtin_amdgcn_wmma_f32_16x16x4_f32(
          /*neg_a=*/false, a, /*neg_b=*/false, b,
          /*c_mod=*/(short)0, acc[s], /*reuse_a=*/false, /*reuse_b=*/false);
    }
  }

  const int rbase = tileM + (lane >> 4) * 8;
#pragma unroll
  for (int s = 0; s < 4; ++s) {
    const int col = ncol0 + s * 16;
    const float rv = rowvec ? rowvec[col] : 0.0f;
#pragma unroll
    for (int r = 0; r < 8; ++r) {
      C[(size_t)(rbase + r) * N + col] = acc[s][r] + rv;
    }
  }
}

// ---------------------------------------------------------------------------
// Aggregation:  g = dinv[i]^2 * h + bias  (self-loop term + bias, init pass)
// ---------------------------------------------------------------------------
__global__ void agg_init(const float* __restrict__ h,
                         const float* __restrict__ dinv,
                         const float* __restrict__ bias,
                         float* __restrict__ g, int n, int fshift) {
  int idx = blockIdx.x * blockDim.x + threadIdx.x;
  int total = n << fshift;
  if (idx >= total) return;
  int i = idx >> fshift;
  int j = idx & ((1 << fshift) - 1);
  float di = dinv[i];
  g[idx] = di * di * h[idx] + bias[j];
}

// One block per edge, one thread per 4-float feature chunk.
__global__ void edge_scatter(const float* __restrict__ h,
                             const long long* __restrict__ src,
                             const long long* __restrict__ dst,
                             const float* __restrict__ dinv,
                             float* __restrict__ g, int F) {
  const int e = blockIdx.x;
  const int c = threadIdx.x * 4;         // blockDim.x == F/4
  const int s = (int)src[e];
  const int d = (int)dst[e];
  const float w = dinv[s] * dinv[d];
  const float4 v = *reinterpret_cast<const float4*>(h + (size_t)s * F + c);
  float* out = g + (size_t)d * F + c;
  atomicAdd(out + 0, v.x * w);
  atomicAdd(out + 1, v.y * w);
  atomicAdd(out + 2, v.z * w);
  atomicAdd(out + 3, v.w * w);
}

__global__ void prelu_k(float* __restrict__ g, const float* __restrict__ a,
                        int total, int fmask) {
  int idx = blockIdx.x * blockDim.x + threadIdx.x;
  if (idx >= total) return;
  float v = g[idx];
  g[idx] = (v >= 0.0f) ? v : a[idx & fmask] * v;
}

// ---------------------------------------------------------------------------
// Launch
// ---------------------------------------------------------------------------
extern "C" void kernel_launch(void* const* d_in, const int* in_sizes, int n_in,
                              void* d_out, int out_size, void* d_ws,
                              size_t ws_size, hipStream_t stream) {
  const float*     x   = (const float*)d_in[0];
  const long long* ei  = (const long long*)d_in[1];   // int64 edge_index [2,E]
  const float*     idv = (const float*)d_in[2];
  const float*     W1  = (const float*)d_in[3];       // 136 x 256
  const float*     b1  = (const float*)d_in[4];
  const float*     a1  = (const float*)d_in[5];
  const float*     W2  = (const float*)d_in[6];       // 256 x 128
  const float*     b2  = (const float*)d_in[7];
  const float*     a2  = (const float*)d_in[8];

  const int N = in_sizes[0] / 128;   // 50000
  const int E = in_sizes[1] / 2;     // 800000
  const long long* src = ei;
  const long long* dst = ei + E;

  // Workspace layout (floats): deg | dinv | c1 | h1 | g1 | h2  (~128.4 MB)
  float* ws   = (float*)d_ws;
  float* deg  = ws;
  float* dinv = deg + N;
  float* c1   = dinv + N;
  float* h1   = c1 + 256;
  float* g1   = h1 + (size_t)N * 256;
  float* h2   = g1 + (size_t)N * 256;
  float* out  = (float*)d_out;

  // Normalization coefficients
  deg_init<<<(N + 255) / 256, 256, 0, stream>>>(deg, N);
  deg_edges<<<(E + 255) / 256, 256, 0, stream>>>(dst, deg, E);
  dinv_k<<<(N + 255) / 256, 256, 0, stream>>>(deg, dinv, N);
  c1_kernel<<<1, 256, 0, stream>>>(idv, W1, c1);

  // Layer 1: h1 = x @ W1[0:128,:] + c1 ; aggregate ; +b1 ; PReLU(a1)
  gemm_f32_wmma<<<dim3((N + 127) / 128, 256 / 64), 256, 0, stream>>>(
      x, W1, c1, h1, N, 256, 128);
  agg_init<<<((N << 8) + 255) / 256, 256, 0, stream>>>(h1, dinv, b1, g1, N, 8);
  edge_scatter<<<E, 256 / 4, 0, stream>>>(h1, src, dst, dinv, g1, 256);
  prelu_k<<<((N << 8) + 255) / 256, 256, 0, stream>>>(g1, a1, N << 8, 255);

  // Layer 2: h2 = g1 @ W2 ; aggregate into d_out ; +b2 ; PReLU(a2)
  gemm_f32_wmma<<<dim3((N + 127) / 128, 128 / 64), 256, 0, stream>>>(
      g1, W2, nullptr, h2, N, 128, 256);
  agg_init<<<((N << 7) + 255) / 256, 256, 0, stream>>>(h2, dinv, b2, out, N, 7);
  edge_scatter<<<E, 128 / 4, 0, stream>>>(h2, src, dst, dinv, out, 128);
  prelu_k<<<((N << 7) + 255) / 256, 256, 0, stream>>>(out, a2, N << 7, 127);
}